// Model_75230647156870
// MI455X (gfx1250) — compile-verified
//
#include <hip/hip_runtime.h>
#include <math.h>

typedef __attribute__((ext_vector_type(2))) float v2f;
typedef __attribute__((ext_vector_type(8))) float v8f;

#define TS        6
#define SRES      256
#define NPIX      (SRES * SRES)
#define NFACES    500
#define PADFACES  512
#define NROWS     (PADFACES * 4)   // 2048 affine rows (w0,w1,w2,z per face)
#define NCHUNKS   (NROWS / 16)     // 128 WMMA M-tiles
#define COEFS_F   (NROWS * 4)      // 8192 floats = 32 KB per view
#define DIST_F    2.732f

// ---------------------------------------------------------------------------
// Kernel 1: per-face prep. Transform vertices by look_at(eye), build the
// per-face affine coefficient rows [Ax, Ay, C, 0] for w0, w1, w2, z with
// 1/area folded in. Degenerate faces (|area| <= 1e-8) get inv = 0 -> all-zero
// rows -> z = 0 -> depth = inf, matching the reference's `ok` masking.
// ---------------------------------------------------------------------------
__global__ void prep_faces(const float* __restrict__ verts,
                           const int*   __restrict__ faces,
                           float*       __restrict__ coefs /* [2][NROWS][4] */)
{
    const int f    = threadIdx.x;   // 0..511
    const int view = blockIdx.y;    // 0: eye=(0,0,+D), 1: eye=(0,0,-D)
    float* out = coefs + (size_t)view * COEFS_F + (size_t)f * 16;

    if (f >= NFACES) {
        #pragma unroll
        for (int k = 0; k < 16; ++k) out[k] = 0.0f;
        return;
    }

    // look_at basis for eye = (0, 0, ez):
    // z = -eye/|eye| = (0,0,-sign(ez)); x = normalize(cross(up,z)) = (z.z,0,0);
    // y = cross(z,x) = (0,1,0)
    const float ez = view ? -DIST_F : DIST_F;
    const float zz = (ez > 0.0f) ? -1.0f : 1.0f;
    const float xx = zz;

    float px[3], py[3], pz[3];
    #pragma unroll
    for (int k = 0; k < 3; ++k) {
        const int vi = faces[f * 3 + k];
        const float vx = verts[vi * 3 + 0];
        const float vy = verts[vi * 3 + 1];
        const float vz = verts[vi * 3 + 2] - ez;
        px[k] = xx * vx;        // row x dot (v - eye)
        py[k] = vy;             // row y dot (v - eye)
        pz[k] = zz * vz;        // row z dot (v - eye)
    }

    // edge(a,b) = (PX-ax)(by-ay) - (PY-ay)(bx-ax) = Ax*PX + Ay*PY + C
    // e0 = edge(p1,p2), e1 = edge(p2,p0), e2 = edge(p0,p1)
    float ax0 = py[2] - py[1], ay0 = -(px[2] - px[1]), c0 = py[1]*px[2] - px[1]*py[2];
    float ax1 = py[0] - py[2], ay1 = -(px[0] - px[2]), c1 = py[2]*px[0] - px[2]*py[0];
    float ax2 = py[1] - py[0], ay2 = -(px[1] - px[0]), c2 = py[0]*px[1] - px[0]*py[1];

    const float area = (px[1]-px[0])*(py[2]-py[0]) - (py[1]-py[0])*(px[2]-px[0]);
    const float inv  = (fabsf(area) > 1e-8f) ? (1.0f / area) : 0.0f;

    ax0 *= inv; ay0 *= inv; c0 *= inv;
    ax1 *= inv; ay1 *= inv; c1 *= inv;
    ax2 *= inv; ay2 *= inv; c2 *= inv;

    // z = w0*z0 + w1*z1 + w2*z2 is affine with component-wise combined coeffs
    const float zax = ax0*pz[0] + ax1*pz[1] + ax2*pz[2];
    const float zay = ay0*pz[0] + ay1*pz[1] + ay2*pz[2];
    const float zc  = c0 *pz[0] + c1 *pz[1] + c2 *pz[2];

    out[ 0] = ax0; out[ 1] = ay0; out[ 2] = c0;  out[ 3] = 0.0f;
    out[ 4] = ax1; out[ 5] = ay1; out[ 6] = c1;  out[ 7] = 0.0f;
    out[ 8] = ax2; out[ 9] = ay2; out[10] = c2;  out[11] = 0.0f;
    out[12] = zax; out[13] = zay; out[14] = zc;  out[15] = 0.0f;
}

// ---------------------------------------------------------------------------
// Kernel 2: WMMA rasterizer. The block first stages the full 32 KB per-view
// coefficient table into LDS via gfx1250 async copies
// (global_load_async_to_lds_b128 + s_wait_asynccnt), then each wave32 owns
// 16 pixels and loops 128 M-tiles: one V_WMMA_F32_16X16X4_F32 per tile
// evaluates (w0,w1,w2,z) for 4 faces x 16 pixels from ds_load_b64 A-frags.
// Branchless per-lane argmin, half-wave shfl_xor(16) merge, texel gather+SSE.
// ---------------------------------------------------------------------------
__global__ void raster_wmma(const float* __restrict__ coefs,
                            const float* __restrict__ tex,   // [NFACES*216][3]
                            const float* __restrict__ ref0,
                            const float* __restrict__ ref1,
                            float*       __restrict__ partials)
{
    __shared__ float lds_coefs[COEFS_F];   // 32 KB
    __shared__ float sred[256];

    const int view = blockIdx.y;
    const float* __restrict__ ref = view ? ref1 : ref0;
    const float* __restrict__ cf  = coefs + (size_t)view * COEFS_F;

    // ---- async stage: 8192 floats = 2048 x b128, 8 per thread, coalesced ----
    {
        const unsigned ldsbase = (unsigned)(uintptr_t)(void*)lds_coefs;
        #pragma unroll
        for (int k = 0; k < 8; ++k) {
            const unsigned loff = ldsbase + threadIdx.x * 16u + (unsigned)k * 4096u;
            const float*   g    = cf + threadIdx.x * 4 + k * 1024;
            asm volatile("global_load_async_to_lds_b128 %0, %1, off"
                         :: "v"(loff), "v"(g) : "memory");
        }
        asm volatile("s_wait_asynccnt 0" ::: "memory");
        __syncthreads();
    }

    const int lane  = threadIdx.x & 31;
    const int wave  = threadIdx.x >> 5;
    const int group = blockIdx.x * 8 + wave;   // 16 pixels per wave
    const int lidx  = lane & 15;
    const bool hi   = lane >= 16;
    const int pix   = group * 16 + lidx;       // this lane's pixel column N
    const int pi    = pix >> 8;                // image row
    const int pj    = pix & (SRES - 1);        // image col

    const float PX =  (2.0f * (float)pj + 1.0f - (float)SRES) * (1.0f / (float)SRES);
    const float PY = -(2.0f * (float)pi + 1.0f - (float)SRES) * (1.0f / (float)SRES);

    // B (4x16): lanes 0-15 carry K=0,1 = (PX, PY); lanes 16-31 carry K=2,3 = (1, 0)
    v2f b;
    b.x = hi ? 1.0f : PX;
    b.y = hi ? 0.0f : PY;

    float bestD = INFINITY;
    int   bestF = 0x7fffffff;
    float bw0 = 0.0f, bw1 = 0.0f, bw2 = 0.0f;

    // A rows from LDS: lanes 0-15 read K=0,1 (Ax,Ay); lanes 16-31 read K=2,3 (C,0)
    // address stride: 4 banks/lane x 16 lanes -> all 64 banks, conflict-free
    const int aoff = hi ? 2 : 0;

    #pragma unroll 4
    for (int ch = 0; ch < NCHUNKS; ++ch) {
        const v2f a = *(const v2f*)&lds_coefs[(ch * 16 + lidx) * 4 + aoff];

        v8f acc = {};
        acc = __builtin_amdgcn_wmma_f32_16x16x4_f32(
            /*neg_a=*/false, a, /*neg_b=*/false, b,
            /*c_mod=*/(short)0, acc, /*reuse_a=*/false, /*reuse_b=*/false);

        // lane<16: rows 0-7 -> faces 4ch+0, 4ch+1 ; lane>=16: rows 8-15 -> +2, +3
        const int f0 = ch * 4 + (hi ? 2 : 0);
        {
            const float w0 = acc[0], w1 = acc[1], w2 = acc[2], z = acc[3];
            const bool in = (w0 >= 0.0f) & (w1 >= 0.0f) & (w2 >= 0.0f) & (z > 0.0f);
            const float d = in ? z : INFINITY;
            const bool take = d < bestD;          // strict < keeps earliest face
            bestD = take ? d  : bestD;
            bestF = take ? f0 : bestF;
            bw0   = take ? w0 : bw0;
            bw1   = take ? w1 : bw1;
            bw2   = take ? w2 : bw2;
        }
        {
            const float w0 = acc[4], w1 = acc[5], w2 = acc[6], z = acc[7];
            const bool in = (w0 >= 0.0f) & (w1 >= 0.0f) & (w2 >= 0.0f) & (z > 0.0f);
            const float d = in ? z : INFINITY;
            const bool take = d < bestD;
            bestD = take ? d      : bestD;
            bestF = take ? f0 + 1 : bestF;
            bw0   = take ? w0     : bw0;
            bw1   = take ? w1     : bw1;
            bw2   = take ? w2     : bw2;
        }
    }

    // merge lane L (faces {0,1} mod 4) with lane L+16 (faces {2,3} mod 4);
    // on depth ties prefer lower face index (argmin first-occurrence)
    const float oD = __shfl_xor(bestD, 16, 32);
    const int   oF = __shfl_xor(bestF, 16, 32);
    const float o0 = __shfl_xor(bw0, 16, 32);
    const float o1 = __shfl_xor(bw1, 16, 32);
    const float o2 = __shfl_xor(bw2, 16, 32);
    {
        const bool take = (oD < bestD) || ((oD == bestD) && (oF < bestF));
        bestD = take ? oD : bestD;
        bestF = take ? oF : bestF;
        bw0   = take ? o0 : bw0;
        bw1   = take ? o1 : bw1;
        bw2   = take ? o2 : bw2;
    }

    float err = 0.0f;
    if (!hi) {
        const bool hit = bestD < INFINITY;
        float c0 = 0.0f, c1 = 0.0f, c2 = 0.0f;
        if (hit && bestF < NFACES) {
            const float s = (float)(TS - 1);
            const int t0 = (int)fminf(fmaxf(rintf(bw0 * s), 0.0f), s);
            const int t1 = (int)fminf(fmaxf(rintf(bw1 * s), 0.0f), s);
            const int t2 = (int)fminf(fmaxf(rintf(bw2 * s), 0.0f), s);
            const int flat = ((bestF * TS + t0) * TS + t1) * TS + t2;
            const float* tp = tex + (size_t)flat * 3;
            c0 = tanhf(tp[0]); c1 = tanhf(tp[1]); c2 = tanhf(tp[2]);
        }
        const float d0 = c0 - ref[0 * NPIX + pix];
        const float d1 = c1 - ref[1 * NPIX + pix];
        const float d2 = c2 - ref[2 * NPIX + pix];
        err = d0 * d0 + d1 * d1 + d2 * d2;
    }

    sred[threadIdx.x] = err;
    __syncthreads();
    #pragma unroll
    for (int s = 128; s > 0; s >>= 1) {
        if (threadIdx.x < s) sred[threadIdx.x] += sred[threadIdx.x + s];
        __syncthreads();
    }
    if (threadIdx.x == 0)
        partials[blockIdx.y * gridDim.x + blockIdx.x] = sred[0];
}

// ---------------------------------------------------------------------------
// Kernel 3: deterministic final reduction of 1024 block partials, /2.
// ---------------------------------------------------------------------------
__global__ void finalize(const float* __restrict__ partials,
                         float*       __restrict__ out)
{
    __shared__ float sred[256];
    const int t = threadIdx.x;
    sred[t] = partials[t] + partials[t + 256] + partials[t + 512] + partials[t + 768];
    __syncthreads();
    #pragma unroll
    for (int s = 128; s > 0; s >>= 1) {
        if (t < s) sred[t] += sred[t + s];
        __syncthreads();
    }
    if (t == 0) out[0] = sred[0] * 0.5f;
}

extern "C" void kernel_launch(void* const* d_in, const int* in_sizes, int n_in,
                              void* d_out, int out_size, void* d_ws, size_t ws_size,
                              hipStream_t stream)
{
    const float* verts = (const float*)d_in[0];   // (252,3) f32
    const int*   faces = (const int*)d_in[1];     // (500,3) int
    const float* tex   = (const float*)d_in[2];   // (500,6,6,6,3) f32
    const float* ref0  = (const float*)d_in[3];   // (3,256,256) f32
    const float* ref1  = (const float*)d_in[4];   // (3,256,256) f32
    float* out = (float*)d_out;

    float* coefs    = (float*)d_ws;                    // 2 * 8192 floats (64 KB)
    float* partials = coefs + (size_t)2 * COEFS_F;     // 1024 floats

    prep_faces<<<dim3(1, 2), PADFACES, 0, stream>>>(verts, faces, coefs);
    raster_wmma<<<dim3(512, 2), 256, 0, stream>>>(coefs, tex, ref0, ref1, partials);
    finalize<<<1, 256, 0, stream>>>(partials, out);
}